// HysteresisRouter_23398981828900
// MI455X (gfx1250) — compile-verified
//
#include <hip/hip_runtime.h>
#include <math.h>

#define NT 16384   // tokens
#define DM 2048    // d_model
#define NE 64      // experts
#define TK 8       // top-k

constexpr float LAMBDA_TAX = 0.04f;
constexpr float RATIO      = (float)NE / (float)NT;   // E/n for Sinkhorn

typedef float v2f __attribute__((ext_vector_type(2)));
typedef float v8f __attribute__((ext_vector_type(8)));

__device__ __forceinline__ float wsum32(float v) {
#pragma unroll
    for (int o = 16; o; o >>= 1) v += __shfl_xor(v, o, 32);
    return v;
}
__device__ __forceinline__ float wmax32(float v) {
#pragma unroll
    for (int o = 16; o; o >>= 1) v = fmaxf(v, __shfl_xor(v, o, 32));
    return v;
}

// One 16-byte async DMA transfer: global -> LDS, tracked by ASYNCcnt.
// LDS operand = low 32 bits of the generic pointer (flat-LDS addr is
// {shared aperture, lds_offset} per the ISA, so truncation gives the offset).
__device__ __forceinline__ void async_copy_b128(const float* gsrc, float* ldst) {
    asm volatile("global_load_async_to_lds_b128 %0, %1, off"
                 :
                 : "v"((unsigned int)(unsigned long long)(uintptr_t)ldst),
                   "v"((unsigned long long)(uintptr_t)gsrc)
                 : "memory");
}

// ---------------------------------------------------------------------------
// Kernel 1: logits = x@W^T + b ; center by row mean ; M = softmax(logits)
// Block = 256 thr (8 waves) -> 32 tokens x 64 experts via 2x4 grid of
// 16x16 V_WMMA_F32_16X16X4_F32 tiles. K staged 64-wide through LDS with
// double-buffered GLOBAL_LOAD_ASYNC_TO_LDS_B128 DMA (6 x b128 per thread
// per chunk; in-order ASYNCcnt lets us overlap next-chunk DMA with WMMA).
// ---------------------------------------------------------------------------
__global__ __launch_bounds__(256) void k_gemm_softmax(
    const float* __restrict__ x, const float* __restrict__ W,
    const float* __restrict__ bias, float* __restrict__ mout,
    float* __restrict__ Lws)
{
    __shared__ __align__(16) float xs[2][32][68];   // x tiles (+4 pad)
    __shared__ __align__(16) float wld[2][64][68];  // W tiles
    __shared__ __align__(16) float lt[32][68];      // logits tile (epilogue)

    const int tid  = threadIdx.x;
    const int wave = tid >> 5, lane = tid & 31;
    const int half = lane >> 4, l16 = lane & 15;
    const int tw = wave >> 2;      // token-tile 0..1
    const int te = wave & 3;       // expert-tile 0..3
    const int tokBase = blockIdx.x * 32;

    // this thread's staging slots (same for every chunk)
    const int rx0 = tid >> 4,          cx0 = (tid & 15) << 2;          // xs part 0
    const int rx1 = (tid + 256) >> 4,  cx1 = cx0;                      // xs part 1
    v8f acc = {0.f, 0.f, 0.f, 0.f, 0.f, 0.f, 0.f, 0.f};

    // issue chunk 0 into buffer 0
    {
        async_copy_b128(&x[(size_t)(tokBase + rx0) * DM + cx0], &xs[0][rx0][cx0]);
        async_copy_b128(&x[(size_t)(tokBase + rx1) * DM + cx1], &xs[0][rx1][cx1]);
#pragma unroll
        for (int i = 0; i < 4; i++) {
            int idx = tid + i * 256, r = idx >> 4, c = (idx & 15) << 2;
            async_copy_b128(&W[(size_t)r * DM + c], &wld[0][r][c]);
        }
    }

    int cur = 0;
    for (int k0 = 0; k0 < DM; k0 += 64) {
        const bool hasNext = (k0 + 64 < DM);
        if (hasNext) {                               // issue chunk k0+64 -> other buf
            const int nb = cur ^ 1, kn = k0 + 64;
            async_copy_b128(&x[(size_t)(tokBase + rx0) * DM + kn + cx0], &xs[nb][rx0][cx0]);
            async_copy_b128(&x[(size_t)(tokBase + rx1) * DM + kn + cx1], &xs[nb][rx1][cx1]);
#pragma unroll
            for (int i = 0; i < 4; i++) {
                int idx = tid + i * 256, r = idx >> 4, c = (idx & 15) << 2;
                async_copy_b128(&W[(size_t)r * DM + kn + c], &wld[nb][r][c]);
            }
            // async loads complete in order: <=6 outstanding => current buf done
            asm volatile("s_wait_asynccnt 0x6" ::: "memory");
        } else {
            asm volatile("s_wait_asynccnt 0x0" ::: "memory");
        }
        __syncthreads();

        // A lane layout (16x4 f32): M=l16, K = kk + 2*half + {0,1}
        // B lane layout (4x16 f32): N=l16, K = kk + 2*half + {0,1}
#pragma unroll
        for (int kk = 0; kk < 64; kk += 4) {
            v2f a  = *(const v2f*)(&xs[cur][16 * tw + l16][kk + 2 * half]);
            v2f bb = *(const v2f*)(&wld[cur][16 * te + l16][kk + 2 * half]);
            acc = __builtin_amdgcn_wmma_f32_16x16x4_f32(
                false, a, false, bb, (short)0, acc, false, false);
        }
        __syncthreads();   // all waves done reading before buf is re-filled
        cur ^= 1;
    }

    // epilogue: bias add, spill 16x16 tile to LDS (row = v + 8*half, col = l16)
    float be = bias[16 * te + l16];
#pragma unroll
    for (int v = 0; v < 8; v++)
        lt[16 * tw + 8 * half + v][16 * te + l16] = acc[v] + be;
    __syncthreads();

    // per-row mean-center + stable softmax: 8 threads x 8 experts per row
    const int r = tid >> 3, g = tid & 7;
    float lv[8];
    float ps = 0.f, pm = -INFINITY;
#pragma unroll
    for (int j = 0; j < 8; j++) {
        lv[j] = lt[r][g * 8 + j];
        ps += lv[j];
        pm = fmaxf(pm, lv[j]);
    }
#pragma unroll
    for (int o = 4; o; o >>= 1) {
        ps += __shfl_xor(ps, o, 8);
        pm = fmaxf(pm, __shfl_xor(pm, o, 8));
    }
    const float mean = ps * (1.f / 64.f);
    float es[8], se = 0.f;
#pragma unroll
    for (int j = 0; j < 8; j++) { es[j] = __expf(lv[j] - pm); se += es[j]; }
#pragma unroll
    for (int o = 4; o; o >>= 1) se += __shfl_xor(se, o, 8);
    const float inv = 1.f / se;
    const size_t base = (size_t)(tokBase + r) * NE + g * 8;
#pragma unroll
    for (int j = 0; j < 8; j++) {
        mout[base + j] = es[j] * inv;      // softmax (shift-invariant, tau=1)
        Lws[base + j]  = lv[j] - mean;     // centered logits kept for stage 3
    }
}

// ---------------------------------------------------------------------------
// Kernel 2: C += M^T M  (64x64). Each block reduces 256 rows; per-thread 4x4
// register sub-block, rows staged 8-at-a-time in LDS, atomicAdd epilogue.
// ---------------------------------------------------------------------------
__global__ __launch_bounds__(256) void k_ctc(const float* __restrict__ M,
                                             float* __restrict__ C)
{
    __shared__ float rows[8][64];
    const int tid = threadIdx.x;
    const int bi = tid >> 4, bj = tid & 15;
    float acc[4][4] = {};
    const int rowBase = blockIdx.x * 256;

    for (int ch = 0; ch < 32; ++ch) {
        if (tid < 128) {
            int r = tid >> 4, c = (tid & 15) << 2;
            *(float4*)(&rows[r][c]) =
                *(const float4*)(&M[(size_t)(rowBase + ch * 8 + r) * NE + c]);
        }
        __syncthreads();
#pragma unroll
        for (int rr = 0; rr < 8; rr++) {
            float mi[4], mj[4];
#pragma unroll
            for (int a = 0; a < 4; a++) { mi[a] = rows[rr][bi * 4 + a]; mj[a] = rows[rr][bj * 4 + a]; }
#pragma unroll
            for (int a = 0; a < 4; a++)
#pragma unroll
                for (int b = 0; b < 4; b++) acc[a][b] += mi[a] * mj[b];
        }
        __syncthreads();
    }
#pragma unroll
    for (int a = 0; a < 4; a++)
#pragma unroll
        for (int b = 0; b < 4; b++)
            atomicAdd(&C[(size_t)(bi * 4 + a) * NE + bj * 4 + b], acc[a][b]);
}

// ---------------------------------------------------------------------------
// Kernel 3: correlation-tax correction + second softmax (in place on M),
// also seeds the first Sinkhorn column sum. One wave per token (2 exp/lane).
// Diagonal of C zeroed at LDS load time (== C * (1-I)).
// ---------------------------------------------------------------------------
__global__ __launch_bounds__(256) void k_correct(
    float* __restrict__ M, const float* __restrict__ Lws,
    const float* __restrict__ C, float* __restrict__ colsum)
{
    __shared__ float Cl[NE * NE];
    __shared__ float mrow[8][64];
    __shared__ float cs[64];
    const int tid = threadIdx.x, wave = tid >> 5, lane = tid & 31;

#pragma unroll
    for (int i = 0; i < 4; i++) {
        int i4 = tid + i * 256;
        *(float4*)(&Cl[i4 * 4]) = *(const float4*)(&C[(size_t)i4 * 4]);
    }
    if (tid < 64) cs[tid] = 0.f;
    __syncthreads();
    if (tid < 64) Cl[tid * 65] = 0.f;   // zero diagonal

    const size_t tok = (size_t)blockIdx.x * 8 + wave;
    const float m0 = M[tok * NE + lane];
    const float m1 = M[tok * NE + lane + 32];
    mrow[wave][lane] = m0;
    mrow[wave][lane + 32] = m1;
    __syncthreads();

    float g0 = 0.f, g1 = 0.f;
    for (int k = 0; k < 64; k++) {
        float mk = mrow[wave][k];
        g0 += mk * Cl[k * 64 + lane];
        g1 += mk * Cl[k * 64 + lane + 32];
    }
    const float gm0 = 4.f * g0, gm1 = 4.f * g1;        // grad_m = 4*(M@C)
    const float dot = wsum32(m0 * gm0 + m1 * gm1);     // sum(M*grad_m)
    const float l20 = Lws[tok * NE + lane]      - LAMBDA_TAX * m0 * (gm0 - dot);
    const float l21 = Lws[tok * NE + lane + 32] - LAMBDA_TAX * m1 * (gm1 - dot);
    const float mx = wmax32(fmaxf(l20, l21));
    const float e0 = __expf(l20 - mx), e1 = __expf(l21 - mx);
    const float s = wsum32(e0 + e1);
    const float p0 = e0 / s, p1 = e1 / s;
    M[tok * NE + lane] = p0;
    M[tok * NE + lane + 32] = p1;
    atomicAdd(&cs[lane], p0);
    atomicAdd(&cs[lane + 32], p1);
    __syncthreads();
    if (tid < 64) atomicAdd(&colsum[tid], cs[tid]);
}

// ---------------------------------------------------------------------------
// Kernel 4: one Sinkhorn iteration: col-scale by (E/n)/clip(colsum), row
// normalize, and accumulate NEXT iteration's column sums (fused).
// ---------------------------------------------------------------------------
__global__ __launch_bounds__(256) void k_sinkhorn(
    float* __restrict__ M, const float* __restrict__ csIn,
    float* __restrict__ csOut)
{
    __shared__ float scale[64];
    __shared__ float csn[64];
    const int tid = threadIdx.x, wave = tid >> 5, lane = tid & 31;
    if (tid < 64) {
        scale[tid] = RATIO / fmaxf(csIn[tid], 1e-12f);
        csn[tid] = 0.f;
    }
    __syncthreads();
    const float s0 = scale[lane], s1 = scale[lane + 32];
    float a0 = 0.f, a1 = 0.f;
    for (int j = 0; j < 8; j++) {
        const size_t tok = (size_t)blockIdx.x * 64 + wave * 8 + j;
        float v0 = M[tok * NE + lane] * s0;
        float v1 = M[tok * NE + lane + 32] * s1;
        const float rs = wsum32(v0 + v1);
        const float inv = 1.f / fmaxf(rs, 1e-12f);
        v0 *= inv; v1 *= inv;
        M[tok * NE + lane] = v0;
        M[tok * NE + lane + 32] = v1;
        a0 += v0; a1 += v1;
    }
    atomicAdd(&csn[lane], a0);
    atomicAdd(&csn[lane + 32], a1);
    __syncthreads();
    if (tid < 64) atomicAdd(&csOut[tid], csn[tid]);
}

// ---------------------------------------------------------------------------
// Kernel 5: damping + hysteresis + renorm + top-8 mask. One wave per token.
// ---------------------------------------------------------------------------
__global__ __launch_bounds__(256) void k_final(
    float* __restrict__ M, const float* __restrict__ pp,
    const unsigned char* __restrict__ pmask, float* __restrict__ maskout)
{
    const int tid = threadIdx.x, wave = tid >> 5, lane = tid & 31;
    const size_t b = ((size_t)blockIdx.x * 8 + wave) * NE;
    const float m0 = M[b + lane],  m1 = M[b + lane + 32];
    const float p0 = pp[b + lane], p1 = pp[b + lane + 32];
    const float f0 = pmask[b + lane] ? 1.f : 0.f;
    const float f1 = pmask[b + lane + 32] ? 1.f : 0.f;
    const float denom = fmaxf(wsum32(f0 + f1), 1.f);
    float v0 = 0.9f * (0.3f * p0 + 0.7f * m0) + 0.1f * (f0 / denom);
    float v1 = 0.9f * (0.3f * p1 + 0.7f * m1) + 0.1f * (f1 / denom);
    const float rs = wsum32(v0 + v1);
    const float inv = 1.f / fmaxf(rs, 1e-12f);
    v0 *= inv; v1 *= inv;
    M[b + lane] = v0;
    M[b + lane + 32] = v1;

    // top-8 via 8 rounds of wave argmax (ties -> lowest index, like top_k)
    bool mk0 = false, mk1 = false;
    for (int it = 0; it < TK; it++) {
        const float c0 = mk0 ? -INFINITY : v0;
        const float c1 = mk1 ? -INFINITY : v1;
        float bv; int bi;
        if (c0 >= c1) { bv = c0; bi = lane; } else { bv = c1; bi = lane + 32; }
#pragma unroll
        for (int o = 16; o; o >>= 1) {
            float ov = __shfl_xor(bv, o, 32);
            int   oi = __shfl_xor(bi, o, 32);
            if (ov > bv || (ov == bv && oi < bi)) { bv = ov; bi = oi; }
        }
        if (bi == lane) mk0 = true;
        else if (bi == lane + 32) mk1 = true;
    }
    maskout[b + lane] = mk0 ? 1.f : 0.f;
    maskout[b + lane + 32] = mk1 ? 1.f : 0.f;
}

// ---------------------------------------------------------------------------
extern "C" void kernel_launch(void* const* d_in, const int* in_sizes, int n_in,
                              void* d_out, int out_size, void* d_ws, size_t ws_size,
                              hipStream_t stream)
{
    (void)in_sizes; (void)n_in; (void)out_size; (void)ws_size;
    const float*         x     = (const float*)d_in[0];
    const float*         W     = (const float*)d_in[1];
    const float*         bias  = (const float*)d_in[2];
    const float*         pp    = (const float*)d_in[3];
    const unsigned char* pmask = (const unsigned char*)d_in[4];

    float* mout    = (float*)d_out;                 // M lives in d_out[0:N*E]
    float* maskout = mout + (size_t)NT * NE;        // mask as 0/1 floats

    float* Lws = (float*)d_ws;                      // centered logits [N,E]
    float* C   = Lws + (size_t)NT * NE;             // 64x64
    float* csA = C + NE * NE;                       // colsum ping
    float* csB = csA + NE;                          // colsum pong

    hipMemsetAsync(C,   0, NE * NE * sizeof(float), stream);
    hipMemsetAsync(csA, 0, NE * sizeof(float), stream);

    k_gemm_softmax<<<NT / 32, 256, 0, stream>>>(x, W, bias, mout, Lws);
    k_ctc<<<NT / 256, 256, 0, stream>>>(mout, C);
    k_correct<<<NT / 8, 256, 0, stream>>>(mout, Lws, C, csA);

    float* cur = csA;
    float* nxt = csB;
    for (int i = 0; i < 10; i++) {
        hipMemsetAsync(nxt, 0, NE * sizeof(float), stream);
        k_sinkhorn<<<NT / 64, 256, 0, stream>>>(mout, cur, nxt);
        float* t = cur; cur = nxt; nxt = t;
    }

    k_final<<<NT / 8, 256, 0, stream>>>(mout, pp, pmask, maskout);
}